// TinyRGATEncoder_30614526885990
// MI455X (gfx1250) — compile-verified
//
#include <hip/hip_runtime.h>
#include <hip/hip_bf16.h>
#include <math.h>

// ---------------- problem constants (match reference) ----------------
#define NN      50000
#define NE      800000
#define DIN     256
#define DH      128
#define NL      2
#define NR      8
#define RDIM    16
#define CONF_W  0.5f
#define LN_EPS  1e-05f

typedef float v2f __attribute__((ext_vector_type(2)));
typedef float v8f __attribute__((ext_vector_type(8)));

#if defined(__HIP_DEVICE_COMPILE__)
#if !__has_builtin(__builtin_amdgcn_wmma_f32_16x16x4_f32)
#error "wmma_f32_16x16x4_f32 builtin missing on gfx1250 device pass"
#endif
#endif

// ---------------- fp32 WMMA GEMM: C = act(A[MxK] @ B[KxDH] + bias) ----------------
// One wave per 16x16 tile; 8 waves/block cover all 128 output columns.
// A 16xK tile is staged in LDS once per block (coalesced), killing the 8x
// redundant A traffic; B (<=128KB weights) is served from L2 across blocks.
// Requires M % 16 == 0, K in {128, 256}.
template <int K>
__global__ void wmma_gemm_f32(const float* __restrict__ A,
                              const float* __restrict__ B,
                              const float* __restrict__ bias,
                              float* __restrict__ C,
                              int do_relu)
{
    constexpr int LDA = K + 4;                 // dword padding -> spread LDS banks
    __shared__ float As[16 * LDA];

    const int tid  = threadIdx.x;
    const int wave = tid >> 5;                 // 0..7
    const int lane = tid & 31;
    const int g    = lane >> 4;                // lane group 0/1
    const int m    = lane & 15;
    const int row0 = blockIdx.x * 16;

    // cooperative, coalesced copy of the 16 x K A-tile into LDS
#pragma unroll
    for (int idx = tid; idx < 16 * K; idx += 256) {
        int r = idx / K;                       // K is a power of two -> shifts
        int c = idx % K;
        As[r * LDA + c] = A[(size_t)(row0 + r) * K + c];
    }
    __syncthreads();

    const int col0 = wave * 16;
    v8f acc = {};
    const float* Am = As + m * LDA + 2 * g;    // A[row0+m][k0 + 2g + {0,1}]
    const float* Bp = B + (size_t)(2 * g) * DH + col0 + m;
#pragma unroll 8
    for (int k0 = 0; k0 < K; k0 += 4) {
        v2f a;
        a.x = Am[k0];
        a.y = Am[k0 + 1];
        v2f b;                                 // B[k0+2g+{0,1}][col0+m]
        b.x = Bp[(size_t)k0 * DH];
        b.y = Bp[(size_t)(k0 + 1) * DH];
        acc = __builtin_amdgcn_wmma_f32_16x16x4_f32(
            /*neg_a=*/false, a, /*neg_b=*/false, b,
            /*c_mod=*/(short)0, acc, /*reuse_a=*/false, /*reuse_b=*/false);
    }
    // D layout: lane(g,m) VGPR v -> C[row0 + v + 8g][col0 + m]
#pragma unroll
    for (int v = 0; v < 8; ++v) {
        int row = row0 + v + 8 * g;
        int col = col0 + m;
        float val = acc[v];
        if (bias) val += bias[col];
        if (do_relu) val = fmaxf(val, 0.0f);
        C[(size_t)row * DH + col] = val;
    }
}

// ---------------- monotonic float<->uint encoding for atomic max ----------------
__device__ __forceinline__ unsigned flip_f32(float f) {
    unsigned u = __float_as_uint(f);
    return (u & 0x80000000u) ? ~u : (u | 0x80000000u);
}
__device__ __forceinline__ float unflip_f32(unsigned u) {
    return __uint_as_float((u & 0x80000000u) ? (u & 0x7FFFFFFFu) : ~u);
}

// ---------------- zero / init scratch per layer ----------------
__global__ void init_layer(float* __restrict__ out, float* __restrict__ ssum,
                           unsigned* __restrict__ menc, int nh, int n)
{
    int i = blockIdx.x * blockDim.x + threadIdx.x;
    if (i < nh) out[i] = 0.0f;
    if (i < n) { ssum[i] = 0.0f; menc[i] = 0u; }   // 0u == encode(lowest)
}

// ---------------- per-node attention scalars: hm . a_d , hm . a_s ----------------
__global__ void node_scores(const float* __restrict__ hm,
                            const float* __restrict__ a_d,
                            const float* __restrict__ a_s,
                            float* __restrict__ dsc, float* __restrict__ ssc, int n)
{
    int node = blockIdx.x * (blockDim.x >> 5) + (threadIdx.x >> 5);
    int lane = threadIdx.x & 31;
    if (node >= n) return;
    float sd = 0.0f, ss = 0.0f;
#pragma unroll
    for (int k = 0; k < DH / 32; ++k) {
        int j = lane + 32 * k;
        float v = hm[(size_t)node * DH + j];
        sd += v * a_d[j];
        ss += v * a_s[j];
    }
#pragma unroll
    for (int off = 16; off > 0; off >>= 1) {
        sd += __shfl_down(sd, off, 32);
        ss += __shfl_down(ss, off, 32);
    }
    if (lane == 0) { dsc[node] = sd; ssc[node] = ss; }
}

// ---------------- tiny per-relation score table: (rel_emb[t] @ W_rel) . a_r ----------------
__global__ void rel_scores(const float* __restrict__ rel_emb_l,   // R x RD
                           const float* __restrict__ W_rel_l,     // RD x H
                           const float* __restrict__ a_r,         // H
                           float* __restrict__ rscore)
{
    int t = threadIdx.x;
    if (t >= NR) return;
    float acc = 0.0f;
    for (int d = 0; d < DH; ++d) {
        float rr = 0.0f;
#pragma unroll
        for (int k = 0; k < RDIM; ++k)
            rr += rel_emb_l[t * RDIM + k] * W_rel_l[k * DH + d];
        acc += rr * a_r[d];
    }
    rscore[t] = acc;
}

// ---------------- edge pass 1: logits + segment max (encoded atomicMax) ----------------
__global__ void edge_logits_max(const int* __restrict__ src, const int* __restrict__ dst,
                                const int* __restrict__ etype, const float* __restrict__ eattr,
                                const float* __restrict__ dsc, const float* __restrict__ ssc,
                                const float* __restrict__ rscore,
                                float* __restrict__ e_out, unsigned* __restrict__ menc, int ne)
{
    int i = blockIdx.x * blockDim.x + threadIdx.x;
    if (i >= ne) return;
    int s = src[i], d = dst[i];
    int t = etype[i]; t = t < 0 ? 0 : (t > NR - 1 ? NR - 1 : t);
    float e = dsc[d] + ssc[s] + rscore[t];
    e = (e > 0.0f) ? e : 0.2f * e;                      // leaky_relu(0.2)
    e += CONF_W * logf(fmaxf(eattr[i], 1e-6f));         // + log-confidence
    e_out[i] = e;
    atomicMax(&menc[d], flip_f32(e));
}

// ---------------- edge pass 2: exp(e - max) + segment sum ----------------
__global__ void edge_exp_sum(const int* __restrict__ dst, const float* __restrict__ e,
                             const unsigned* __restrict__ menc,
                             float* __restrict__ ex, float* __restrict__ ssum, int ne)
{
    int i = blockIdx.x * blockDim.x + threadIdx.x;
    if (i >= ne) return;
    int d = dst[i];
    float v = expf(e[i] - unflip_f32(menc[d]));
    ex[i] = v;
    atomicAdd(&ssum[d], v);
}

// ---------------- edge pass 3: out[dst] += alpha * hm[src]  (wave32 per edge) ----------------
__global__ void edge_aggregate(const int* __restrict__ src, const int* __restrict__ dst,
                               const float* __restrict__ ex, const float* __restrict__ ssum,
                               const float* __restrict__ hm, float* __restrict__ out, int ne)
{
    int edge = blockIdx.x * (blockDim.x >> 5) + (threadIdx.x >> 5);
    int lane = threadIdx.x & 31;
    if (edge >= ne) return;
    int s = src[edge], d = dst[edge];
    float alpha = ex[edge] / ssum[d];
    const float* hs = hm + (size_t)s * DH;
    float* od = out + (size_t)d * DH;
#pragma unroll
    for (int k = 0; k < DH / 32; ++k) {
        int j = lane + 32 * k;
        atomicAdd(&od[j], hs[j] * alpha);
    }
}

// ---------------- finalize: h = LayerNorm(h + relu(out + bias)) (wave32 per node) ----------------
__global__ void finalize_ln(const float* __restrict__ h_in, const float* __restrict__ out,
                            const float* __restrict__ bias_l, const float* __restrict__ ln_g,
                            const float* __restrict__ ln_b, float* __restrict__ h_out, int n)
{
    int node = blockIdx.x * (blockDim.x >> 5) + (threadIdx.x >> 5);
    int lane = threadIdx.x & 31;
    if (node >= n) return;
    float vals[DH / 32];
    float sum = 0.0f;
#pragma unroll
    for (int k = 0; k < DH / 32; ++k) {
        int j = lane + 32 * k;
        float v = h_in[(size_t)node * DH + j] + fmaxf(out[(size_t)node * DH + j] + bias_l[j], 0.0f);
        vals[k] = v;
        sum += v;
    }
#pragma unroll
    for (int off = 16; off > 0; off >>= 1) sum += __shfl_down(sum, off, 32);
    sum = __shfl(sum, 0, 32);
    float mu = sum * (1.0f / DH);
    float var = 0.0f;
#pragma unroll
    for (int k = 0; k < DH / 32; ++k) { float dlt = vals[k] - mu; var += dlt * dlt; }
#pragma unroll
    for (int off = 16; off > 0; off >>= 1) var += __shfl_down(var, off, 32);
    var = __shfl(var, 0, 32);
    float inv = rsqrtf(var * (1.0f / DH) + LN_EPS);
#pragma unroll
    for (int k = 0; k < DH / 32; ++k) {
        int j = lane + 32 * k;
        h_out[(size_t)node * DH + j] = (vals[k] - mu) * inv * ln_g[j] + ln_b[j];
    }
}

// ---------------- host driver ----------------
extern "C" void kernel_launch(void* const* d_in, const int* in_sizes, int n_in,
                              void* d_out, int out_size, void* d_ws, size_t ws_size,
                              hipStream_t stream)
{
    const float* x       = (const float*)d_in[0];
    const int*   eidx    = (const int*)  d_in[1];   // [2, E]
    const int*   etype   = (const int*)  d_in[2];   // [E]
    const float* eattr   = (const float*)d_in[3];   // [E, 1]
    const float* W_in    = (const float*)d_in[4];   // [IN, H]
    const float* b_in    = (const float*)d_in[5];   // [H]
    const float* W_msg   = (const float*)d_in[6];   // [L, H, H]
    const float* rel_emb = (const float*)d_in[7];   // [L, R, RD]
    const float* W_rel   = (const float*)d_in[8];   // [L, RD, H]
    const float* att     = (const float*)d_in[9];   // [L, 3H]
    const float* bias    = (const float*)d_in[10];  // [L, H]
    const float* ln_g    = (const float*)d_in[11];  // [L, H]
    const float* ln_b    = (const float*)d_in[12];  // [L, H]

    const int* src = eidx;
    const int* dst = eidx + NE;

    // workspace carve-up (floats)
    float* ws   = (float*)d_ws;
    float* h    = ws;                       // N*H
    float* hm   = h + (size_t)NN * DH;      // N*H
    float* out  = hm + (size_t)NN * DH;     // N*H
    float* dsc  = out + (size_t)NN * DH;    // N
    float* ssc  = dsc + NN;                 // N
    float* ssum = ssc + NN;                 // N
    unsigned* menc = (unsigned*)(ssum + NN);// N
    float* e    = (float*)(menc + NN);      // E
    float* ex   = e + NE;                   // E
    float* rsc  = ex + NE;                  // R

    const int NH = NN * DH;

    // h = relu(x @ W_in + b_in)   -- fp32 WMMA, LDS-staged A tile
    wmma_gemm_f32<DIN><<<NN / 16, 256, 0, stream>>>(x, W_in, b_in, h, 1);

    for (int l = 0; l < NL; ++l) {
        const float* Wm  = W_msg + (size_t)l * DH * DH;
        const float* a_d = att + (size_t)l * 3 * DH;
        const float* a_s = a_d + DH;
        const float* a_r = a_d + 2 * DH;

        // hm = h @ W_msg[l]   -- fp32 WMMA
        wmma_gemm_f32<DH><<<NN / 16, 256, 0, stream>>>(h, Wm, nullptr, hm, 0);

        init_layer<<<(NH + 255) / 256, 256, 0, stream>>>(out, ssum, menc, NH, NN);

        node_scores<<<(NN + 7) / 8, 256, 0, stream>>>(hm, a_d, a_s, dsc, ssc, NN);

        rel_scores<<<1, 32, 0, stream>>>(rel_emb + (size_t)l * NR * RDIM,
                                         W_rel + (size_t)l * RDIM * DH, a_r, rsc);

        edge_logits_max<<<(NE + 255) / 256, 256, 0, stream>>>(src, dst, etype, eattr,
                                                              dsc, ssc, rsc, e, menc, NE);

        edge_exp_sum<<<(NE + 255) / 256, 256, 0, stream>>>(dst, e, menc, ex, ssum, NE);

        edge_aggregate<<<(NE + 7) / 8, 256, 0, stream>>>(src, dst, ex, ssum, hm, out, NE);

        float* h_next = (l == NL - 1) ? (float*)d_out : h;
        finalize_ln<<<(NN + 7) / 8, 256, 0, stream>>>(h, out, bias + (size_t)l * DH,
                                                      ln_g + (size_t)l * DH, ln_b + (size_t)l * DH,
                                                      h_next, NN);
    }
}